// Sub_Cluster_Level_GCN_63745904607644
// MI455X (gfx1250) — compile-verified
//
#include <hip/hip_runtime.h>
#include <hip/hip_bf16.h>

typedef __attribute__((ext_vector_type(2))) float v2f;
typedef __attribute__((ext_vector_type(8))) float v8f;

#define N_      32768
#define D_      2048
#define S_      4096
#define B_      64
#define K_      64
#define NHID_   512
#define NCLASS_ 2
#define TOPK_   5

// ---------------------------------------------------------------------------
// 1) segment counts + segment sums (atomic, coalesced along D)
// ---------------------------------------------------------------------------
__global__ __launch_bounds__(256) void seg_count_kernel(const int* __restrict__ sub_label,
                                                        float* __restrict__ nums) {
    int n = blockIdx.x * blockDim.x + threadIdx.x;
    if (n < N_) atomicAdd(&nums[sub_label[n]], 1.0f);
}

__global__ __launch_bounds__(256) void seg_sum_kernel(const float* __restrict__ feat,
                                                      const int* __restrict__ sub_label,
                                                      float* __restrict__ sub_sum) {
    long gid = (long)blockIdx.x * blockDim.x + threadIdx.x;   // N*D threads
    int d = (int)(gid & (D_ - 1));
    int n = (int)(gid >> 11);
    atomicAdd(&sub_sum[(long)sub_label[n] * D_ + d], feat[gid]);
}

// ---------------------------------------------------------------------------
// 2) sub_sum /= max(nums,1); simm[s] = sum(sub_sum[s]^2)
// ---------------------------------------------------------------------------
__global__ __launch_bounds__(256) void finalize_subsum_kernel(float* __restrict__ sub_sum,
                                                              const float* __restrict__ nums,
                                                              float* __restrict__ simm) {
    int s = blockIdx.x;
    float cnt = nums[s];
    float inv = 1.0f / (cnt > 0.0f ? cnt : 1.0f);
    float acc = 0.0f;
    for (int d = threadIdx.x; d < D_; d += blockDim.x) {
        float v = sub_sum[(long)s * D_ + d] * inv;
        sub_sum[(long)s * D_ + d] = v;
        acc += v * v;
    }
    __shared__ float red[256];
    red[threadIdx.x] = acc;
    __syncthreads();
    for (int off = 128; off > 0; off >>= 1) {
        if (threadIdx.x < off) red[threadIdx.x] += red[threadIdx.x + off];
        __syncthreads();
    }
    if (threadIdx.x == 0) simm[s] = red[0];
}

// ---------------------------------------------------------------------------
// 3) sub_feat[b,k,d] = (k==0) ? features[indexes[b],d] : sub_sum[sub_label[knn[b,k]],d]
// ---------------------------------------------------------------------------
__global__ __launch_bounds__(256) void build_subfeat_kernel(const float* __restrict__ features,
                                                            const float* __restrict__ sub_sum,
                                                            const int* __restrict__ sub_label,
                                                            const int* __restrict__ indexes,
                                                            const int* __restrict__ knn,
                                                            float* __restrict__ sf) {
    long gid = (long)blockIdx.x * blockDim.x + threadIdx.x;   // B*K*D threads
    int d  = (int)(gid & (D_ - 1));
    int bk = (int)(gid >> 11);
    int k  = bk & (K_ - 1);
    int b  = bk >> 6;
    float v;
    if (k == 0) v = features[(long)indexes[b] * D_ + d];
    else        v = sub_sum[(long)sub_label[knn[bk]] * D_ + d];
    sf[gid] = v;
}

// ---------------------------------------------------------------------------
// 4) Generic batched WMMA f32 GEMM:  Out = act( scale*(X0*W0 [+ X1*W1]) + bias )
//    64x64 output tile per block (256 thr = 8 waves, 2 tiles/wave), BK=32 via LDS.
//    wTrans=1 reads W as W[k][n] = Wmem[n*ldw + k] (for SF*SF^T).
// ---------------------------------------------------------------------------
#define BM 64
#define BN 64
#define BKK 32

__global__ __launch_bounds__(256)
void gemm_wmma_kernel(const float* __restrict__ X0, long xbs0, int ldx0,
                      const float* __restrict__ W0, long wbs0, int ldw0,
                      const float* __restrict__ X1, const float* __restrict__ W1,
                      int ldx1, int ldw1, int wTrans,
                      const float* __restrict__ bias, const float* __restrict__ prelu,
                      float* __restrict__ Out, long obs, int ldo,
                      int Kdim, int act, float scale) {
    __shared__ float Xs[BM][36];   // pad 36: conflict-free column reads
    __shared__ float Ws[BKK][68];  // pad 68

    const int bz  = blockIdx.z;
    const int gm0 = blockIdx.y * BM;
    const int gn0 = blockIdx.x * BN;
    const int t    = threadIdx.x;
    const int lane = t & 31;
    const int wv   = t >> 5;          // wave 0..7
    const int tm   = wv >> 1;         // tile row 0..3
    const int tn0  = (wv & 1) * 2;    // tile col base 0 or 2
    const int half = lane >> 4;       // 0: K=0,1  1: K=2,3
    const int r    = lane & 15;

    const float* Xp = X0 + (long)bz * xbs0;
    const float* Wp = W0 + (long)bz * wbs0;
    int ldx = ldx0, ldw = ldw0;

    v8f accA = {0.f,0.f,0.f,0.f,0.f,0.f,0.f,0.f};
    v8f accB = accA;

    for (int pass = 0; pass < 2; ++pass) {
        if (pass == 1) {
            if (!X1) break;
            Xp = X1; Wp = W1; ldx = ldx1; ldw = ldw1;
        }
        for (int kk = 0; kk < Kdim; kk += BKK) {
            __syncthreads();
            {   // stage X tile 64x32: thread -> row t/4, 8 consecutive cols
                int row = t >> 2, cc = (t & 3) * 8;
                const float* src = Xp + (long)(gm0 + row) * ldx + kk + cc;
                #pragma unroll
                for (int j = 0; j < 8; ++j) Xs[row][cc + j] = src[j];
            }
            {   // stage W tile 32x64
                int row = t >> 3, cc = (t & 7) * 8;
                if (!wTrans) {
                    const float* src = Wp + (long)(kk + row) * ldw + gn0 + cc;
                    #pragma unroll
                    for (int j = 0; j < 8; ++j) Ws[row][cc + j] = src[j];
                } else {
                    #pragma unroll
                    for (int j = 0; j < 8; ++j)
                        Ws[row][cc + j] = Wp[(long)(gn0 + cc + j) * ldw + kk + row];
                }
            }
            __syncthreads();
            #pragma unroll
            for (int ks = 0; ks < BKK; ks += 4) {
                int kc = ks + 2 * half;   // ISA f32 A/B layout: lanes16-31 hold K=2,3
                v2f fa, fb0, fb1;
                fa.x  = Xs[tm * 16 + r][kc];
                fa.y  = Xs[tm * 16 + r][kc + 1];
                fb0.x = Ws[kc][tn0 * 16 + r];
                fb0.y = Ws[kc + 1][tn0 * 16 + r];
                fb1.x = Ws[kc][tn0 * 16 + 16 + r];
                fb1.y = Ws[kc + 1][tn0 * 16 + 16 + r];
                accA = __builtin_amdgcn_wmma_f32_16x16x4_f32(false, fa, false, fb0,
                                                             (short)0, accA, false, false);
                accB = __builtin_amdgcn_wmma_f32_16x16x4_f32(false, fa, false, fb1,
                                                             (short)0, accB, false, false);
            }
        }
    }

    float* outp = Out + (long)bz * obs;
    #pragma unroll
    for (int v = 0; v < 8; ++v) {       // C layout: VGPR v -> row v (lanes<16) / v+8
        int row = gm0 + tm * 16 + v + 8 * half;
        #pragma unroll
        for (int j = 0; j < 2; ++j) {
            int col = gn0 + (tn0 + j) * 16 + r;
            float val = ((j == 0) ? accA[v] : accB[v]) * scale;
            if (bias) val += bias[col];
            if (act == 1)      val = fmaxf(val, 0.0f);
            else if (act == 2) val = (val > 0.0f) ? val : prelu[col] * val;
            outp[(long)row * ldo + col] = val;
        }
    }
}

// ---------------------------------------------------------------------------
// 5) per-row top-5 bitmask (strict >, lowest index wins ties == lax.top_k)
// ---------------------------------------------------------------------------
__global__ __launch_bounds__(256) void topk_mask_kernel(const float* __restrict__ A,
                                                        unsigned long long* __restrict__ rowmask) {
    int rid = blockIdx.x * blockDim.x + threadIdx.x;   // B*K rows
    if (rid >= B_ * K_) return;
    const float* row = A + (long)rid * K_;
    unsigned long long m = 0ull;
    for (int tpick = 0; tpick < TOPK_; ++tpick) {
        float best = -3.4e38f;
        int bi = 0;
        for (int l = 0; l < K_; ++l) {
            if (!((m >> l) & 1ull)) {
                float v = row[l];
                if (v > best) { best = v; bi = l; }
            }
        }
        m |= (1ull << bi);
    }
    rowmask[rid] = m;
}

// 6) A *= (mask & mask^T)
__global__ __launch_bounds__(256) void apply_mask_kernel(float* __restrict__ A,
                                                         const unsigned long long* __restrict__ rowmask) {
    int gid = blockIdx.x * blockDim.x + threadIdx.x;   // B*K*K
    int l  = gid & (K_ - 1);
    int bk = gid >> 6;
    int k  = bk & (K_ - 1);
    int b  = bk >> 6;
    bool keep = ((rowmask[bk] >> l) & 1ull) && ((rowmask[b * K_ + l] >> k) & 1ull);
    if (!keep) A[gid] = 0.0f;
}

// 7) row L2 norms (one wave per row)
__global__ __launch_bounds__(256) void row_norms_kernel(const float* __restrict__ sf,
                                                        float* __restrict__ norms) {
    int rid  = blockIdx.x * 8 + (threadIdx.x >> 5);
    int lane = threadIdx.x & 31;
    const float* row = sf + (long)rid * D_;
    float acc = 0.0f;
    for (int d = lane; d < D_; d += 32) { float v = row[d]; acc += v * v; }
    for (int off = 16; off > 0; off >>= 1) acc += __shfl_xor(acc, off, 32);
    if (lane == 0) norms[rid] = sqrtf(acc);
}

// 8) sf[b,k,:] = sf[b,k,:]/||.|| - sf[b,0,:]/||.||  (one thread owns one (b,d) column)
__global__ __launch_bounds__(256) void normalize_center_kernel(float* __restrict__ sf,
                                                               const float* __restrict__ norms) {
    int gid = blockIdx.x * blockDim.x + threadIdx.x;   // B*D
    int d = gid & (D_ - 1);
    int b = gid >> 11;
    float* base = sf + (long)b * K_ * D_ + d;
    float n0 = norms[b * K_];
    float s0 = base[0] / n0;
    for (int k = 0; k < K_; ++k) {
        float v = base[(long)k * D_] / norms[b * K_ + k] - s0;
        base[(long)k * D_] = v;
    }
}

// 9) logits = z @ w2 + b2 ; softmax over 2 classes (one wave per row)
__global__ __launch_bounds__(256) void head_kernel(const float* __restrict__ z,
                                                   const float* __restrict__ w2,
                                                   const float* __restrict__ b2,
                                                   float* __restrict__ pred) {
    int rid  = blockIdx.x * 8 + (threadIdx.x >> 5);
    int lane = threadIdx.x & 31;
    const float* row = z + (long)rid * NHID_;
    float a0 = 0.0f, a1 = 0.0f;
    for (int j = lane; j < NHID_; j += 32) {
        float v = row[j];
        a0 += v * w2[j * 2 + 0];
        a1 += v * w2[j * 2 + 1];
    }
    for (int off = 16; off > 0; off >>= 1) {
        a0 += __shfl_xor(a0, off, 32);
        a1 += __shfl_xor(a1, off, 32);
    }
    if (lane == 0) {
        a0 += b2[0]; a1 += b2[1];
        float m  = fmaxf(a0, a1);
        float e0 = __expf(a0 - m), e1 = __expf(a1 - m);
        float inv = 1.0f / (e0 + e1);
        pred[rid * 2 + 0] = e0 * inv;
        pred[rid * 2 + 1] = e1 * inv;
    }
}

// ---------------------------------------------------------------------------
extern "C" void kernel_launch(void* const* d_in, const int* in_sizes, int n_in,
                              void* d_out, int out_size, void* d_ws, size_t ws_size,
                              hipStream_t stream) {
    const int*   indexes   = (const int*)d_in[0];
    const float* features  = (const float*)d_in[1];
    const int*   sub_label = (const int*)d_in[3];
    const int*   knn       = (const int*)d_in[6];
    const float* conv_w    = (const float*)d_in[8];
    const float* conv_b    = (const float*)d_in[9];
    const float* w1        = (const float*)d_in[10];
    const float* b1        = (const float*)d_in[11];
    const float* prelu_a   = (const float*)d_in[12];
    const float* w2        = (const float*)d_in[13];
    const float* b2        = (const float*)d_in[14];

    // output layout: pred(8192) | simm(4096) | sub_sum(8388608) | nums(4096)
    float* out     = (float*)d_out;
    float* pred    = out;
    float* simm    = out + 8192;
    float* sub_sum = out + 12288;
    float* nums    = out + 12288 + (long)S_ * D_;

    // workspace layout (floats)
    float* ws   = (float*)d_ws;
    float* sf   = ws;                         // B*K*D   = 8388608
    float* agg  = ws + 8388608;               // B*K*D   = 8388608
    float* Amat = ws + 16777216;              // B*K*K   = 262144
    float* hbuf = ws + 17039360;              // 4096*512 = 2097152
    float* zbuf = ws + 19136512;              // 4096*512 = 2097152
    float* norms = ws + 21233664;             // 4096
    unsigned long long* rowmask = (unsigned long long*)(ws + 21237760); // 4096 u64

    // zero outputs (sub_sum/nums accumulated in place; harness poisons d_out)
    hipMemsetAsync(d_out, 0, (size_t)out_size * sizeof(float), stream);

    seg_count_kernel<<<N_ / 256, 256, 0, stream>>>(sub_label, nums);
    seg_sum_kernel<<<(int)(((long)N_ * D_) / 256), 256, 0, stream>>>(features, sub_label, sub_sum);
    finalize_subsum_kernel<<<S_, 256, 0, stream>>>(sub_sum, nums, simm);
    build_subfeat_kernel<<<(B_ * K_ * D_) / 256, 256, 0, stream>>>(features, sub_sum, sub_label,
                                                                   indexes, knn, sf);
    // A[b] = sf[b] * sf[b]^T / 5          (M=64, N=64, K=2048)
    {
        dim3 g(1, 1, B_);
        gemm_wmma_kernel<<<g, 256, 0, stream>>>(sf, (long)K_ * D_, D_,
                                                sf, (long)K_ * D_, D_,
                                                nullptr, nullptr, 0, 0,
                                                /*wTrans=*/1, nullptr, nullptr,
                                                Amat, (long)K_ * K_, K_,
                                                D_, /*act=*/0, 0.2f);
    }
    topk_mask_kernel<<<(B_ * K_) / 256, 256, 0, stream>>>(Amat, rowmask);
    apply_mask_kernel<<<(B_ * K_ * K_) / 256, 256, 0, stream>>>(Amat, rowmask);
    row_norms_kernel<<<(B_ * K_) / 8, 256, 0, stream>>>(sf, norms);
    normalize_center_kernel<<<(B_ * D_) / 256, 256, 0, stream>>>(sf, norms);
    // agg[b] = A[b] * sf[b]               (M=64, N=2048, K=64)
    {
        dim3 g(D_ / 64, 1, B_);
        gemm_wmma_kernel<<<g, 256, 0, stream>>>(Amat, (long)K_ * K_, K_,
                                                sf, (long)K_ * D_, D_,
                                                nullptr, nullptr, 0, 0,
                                                0, nullptr, nullptr,
                                                agg, (long)K_ * D_, D_,
                                                K_, 0, 1.0f);
    }
    // h = relu(sf @ Wtop + agg @ Wbot + conv_b)   (M=4096, N=512, K=2048 x2)
    {
        dim3 g(NHID_ / 64, (B_ * K_) / 64, 1);
        gemm_wmma_kernel<<<g, 256, 0, stream>>>(sf, 0, D_,
                                                conv_w, 0, NHID_,
                                                agg, conv_w + (long)D_ * NHID_, D_, NHID_,
                                                0, conv_b, nullptr,
                                                hbuf, 0, NHID_,
                                                D_, /*relu*/1, 1.0f);
    }
    // z = prelu(h @ w1 + b1)              (M=4096, N=512, K=512)
    {
        dim3 g(NHID_ / 64, (B_ * K_) / 64, 1);
        gemm_wmma_kernel<<<g, 256, 0, stream>>>(hbuf, 0, NHID_,
                                                w1, 0, NHID_,
                                                nullptr, nullptr, 0, 0,
                                                0, b1, prelu_a,
                                                zbuf, 0, NHID_,
                                                NHID_, /*prelu*/2, 1.0f);
    }
    head_kernel<<<(B_ * K_) / 8, 256, 0, stream>>>(zbuf, w2, b2, pred);
}